// Policy_12275016532374
// MI455X (gfx1250) — compile-verified
//
#include <hip/hip_runtime.h>

// CDNA5 / gfx1250, wave32. Full-f32 MLP using V_WMMA_F32_16X16X4_F32.
typedef __attribute__((ext_vector_type(2))) float v2f;  // A/B fragment: 16x4 f32 = 2 VGPRs
typedef __attribute__((ext_vector_type(8))) float v8f;  // C/D fragment: 16x16 f32 = 8 VGPRs

#define LDS_STRIDE       34   // dwords per activation row; padding => conflict-free store+load
#define WAVES_PER_BLOCK   8
#define ROWS_PER_WAVE    32   // two 16-row WMMA tiles per wave (reuses B operands 2x)

__global__ __launch_bounds__(256) void policy_mlp_kernel(
    const float* __restrict__ state,  // [B,32]
    const float* __restrict__ Wh,     // [L,32,32]  (torch Linear: W[out][in])
    const float* __restrict__ bh,     // [L,32]
    const float* __restrict__ Wout,   // [1,32]
    const float* __restrict__ bout,   // [1]
    float*       __restrict__ out,    // [B,1]
    int L)
{
    __shared__ float lds[WAVES_PER_BLOCK * ROWS_PER_WAVE * LDS_STRIDE];

    const int lane = threadIdx.x & 31;
    const int wave = threadIdx.x >> 5;
    const int m    = lane & 15;   // A: row in tile / B: out-neuron in tile / D: column
    const int half = lane >> 4;   // selects which K-pair (A/B) or row-half (D) this lane holds
    const int c0   = 2 * half;    // first of the two K columns per fragment

    const int row0 = (blockIdx.x * WAVES_PER_BLOCK + wave) * ROWS_PER_WAVE;
    float* sw = lds + wave * ROWS_PER_WAVE * LDS_STRIDE;   // per-wave private scratch

    // ---- load two 16x32 activation tiles directly in A-fragment layout ----
    // A 16x4 f32 layout: lane(m,half): VGPR0=A[m][4k+2h], VGPR1=A[m][4k+2h+1] -> one b64 load
    v2f a0[8], a1[8];
    #pragma unroll
    for (int k = 0; k < 8; ++k) {
        a0[k] = *(const v2f*)(state + (size_t)(row0 +  0 + m) * 32 + 4*k + c0);
        a1[k] = *(const v2f*)(state + (size_t)(row0 + 16 + m) * 32 + 4*k + c0);
    }

    // ---- 40 hidden layers: x = relu(x @ W^T + b), all on the f32 matrix pipe ----
    for (int l = 0; l < L; ++l) {
        const float* W = Wh + (size_t)l * (32 * 32);
        v8f acc00 = {}, acc01 = {}, acc10 = {}, acc11 = {};
        #pragma unroll
        for (int k = 0; k < 8; ++k) {
            // B operand B[kk][n] = W[n][kk]; lane(n,half) holds W[n][4k+2h..+1] -> b64 load.
            // Weights are 165KB total: L2/WGP$-resident, broadcast across all waves.
            v2f b0 = *(const v2f*)(W + (size_t)( 0 + m) * 32 + 4*k + c0);  // n = 0..15
            v2f b1 = *(const v2f*)(W + (size_t)(16 + m) * 32 + 4*k + c0);  // n = 16..31
            acc00 = __builtin_amdgcn_wmma_f32_16x16x4_f32(false, a0[k], false, b0, (short)0, acc00, false, false);
            acc01 = __builtin_amdgcn_wmma_f32_16x16x4_f32(false, a0[k], false, b1, (short)0, acc01, false, false);
            acc10 = __builtin_amdgcn_wmma_f32_16x16x4_f32(false, a1[k], false, b0, (short)0, acc10, false, false);
            acc11 = __builtin_amdgcn_wmma_f32_16x16x4_f32(false, a1[k], false, b1, (short)0, acc11, false, false);
        }

        // bias + ReLU, scatter D-layout -> row-major LDS (stride 34: conflict-free)
        const float bias0 = bh[l * 32 +      m];
        const float bias1 = bh[l * 32 + 16 + m];
        #pragma unroll
        for (int r = 0; r < 8; ++r) {
            const int rr = r + 8 * half;             // D: VGPR r is row r+8*half, col = lane
            sw[( 0 + rr) * LDS_STRIDE +      m] = fmaxf(acc00[r] + bias0, 0.0f);
            sw[( 0 + rr) * LDS_STRIDE + 16 + m] = fmaxf(acc01[r] + bias1, 0.0f);
            sw[(16 + rr) * LDS_STRIDE +      m] = fmaxf(acc10[r] + bias0, 0.0f);
            sw[(16 + rr) * LDS_STRIDE + 16 + m] = fmaxf(acc11[r] + bias1, 0.0f);
        }
        __syncthreads();  // order stores -> gathers (scratch is per-wave private)

        // gather row-major LDS -> A-fragment layout for the next layer (b64 ds loads)
        #pragma unroll
        for (int k = 0; k < 8; ++k) {
            a0[k] = *(const v2f*)(sw + ( 0 + m) * LDS_STRIDE + 4*k + c0);
            a1[k] = *(const v2f*)(sw + (16 + m) * LDS_STRIDE + 4*k + c0);
        }
    }

    // ---- head: Linear(32,1). Each lane owns one of this wave's 32 rows. ----
    float accf = bout[0];
    #pragma unroll
    for (int c = 0; c < 32; ++c)
        accf += sw[lane * LDS_STRIDE + c] * Wout[c];   // Wout uniform -> scalar loads
    out[row0 + lane] = accf;
}

extern "C" void kernel_launch(void* const* d_in, const int* in_sizes, int n_in,
                              void* d_out, int out_size, void* d_ws, size_t ws_size,
                              hipStream_t stream) {
    const float* state = (const float*)d_in[0];
    const float* Wh    = (const float*)d_in[1];
    const float* bh    = (const float*)d_in[2];
    const float* Wout  = (const float*)d_in[3];
    const float* bout  = (const float*)d_in[4];
    float* out = (float*)d_out;

    const int B = in_sizes[0] / 32;          // 131072
    const int L = in_sizes[1] / (32 * 32);   // 40

    const int rows_per_block = WAVES_PER_BLOCK * ROWS_PER_WAVE;  // 256
    const int grid = (B + rows_per_block - 1) / rows_per_block;  // 512
    policy_mlp_kernel<<<grid, 256, 0, stream>>>(state, Wh, bh, Wout, bout, out, L);
}